// AlignmentWithRAT_22170621182337
// MI455X (gfx1250) — compile-verified
//
#include <hip/hip_runtime.h>
#include <math.h>

typedef __attribute__((ext_vector_type(2))) float v2f;
typedef __attribute__((ext_vector_type(8))) float v8f;

#define H 128
#define LQ 512
#define LC 384
#define LT 128
#define LTOT 1024           // LQ+LC+LT
#define LKV  512            // LC+LT
#define NREL 51

static __device__ __forceinline__ v8f wmma_f32(v2f a, v2f b, v8f c) {
  return __builtin_amdgcn_wmma_f32_16x16x4_f32(false, a, false, b, (short)0, c,
                                               false, false);
}

// Row pointer into the virtual [enc(1024) ; kv(512)] x 128 input matrix.
static __device__ __forceinline__ const float* row_ptr(int v, const float* qe,
                                                       const float* ce,
                                                       const float* te) {
  if (v < LQ) return qe + v * H;
  if (v < LQ + LC) return ce + (v - LQ) * H;
  if (v < LTOT) return te + (v - LQ - LC) * H;
  v -= LTOT;                      // kv row
  if (v < LC) return ce + v * H;
  return te + (v - LC) * H;
}

// ---------------------------------------------------------------------------
// Kernel 1: Q = enc@Wq+bq (rows 0..1023), K = [c;t]@Wk+bk (rows 1024..1535)
// One block = 16 output rows, 128 threads = 4 waves, each wave does 2 N-tiles.
// ---------------------------------------------------------------------------
__global__ void __launch_bounds__(128)
proj_kernel(const float* __restrict__ qe, const float* __restrict__ ce,
            const float* __restrict__ te, const float* __restrict__ Wq,
            const float* __restrict__ bq, const float* __restrict__ Wk,
            const float* __restrict__ bk, float* __restrict__ Qws,
            float* __restrict__ Kws) {
  __shared__ __align__(16) float as[16 * 132];   // 16x128 A tile, padded rows

  const int v0 = blockIdx.x * 16;
  const int tid = threadIdx.x;
  const int lane = tid & 31;
  const int wave = tid >> 5;
  const int m = lane & 15;
  const int hi = lane >> 4;

  // Stage A tile (coalesced float4 per row).
  for (int idx = tid; idx < 16 * 32; idx += 128) {
    const int r = idx >> 5, c4 = idx & 31;
    const float4 v = ((const float4*)row_ptr(v0 + r, qe, ce, te))[c4];
    *(float4*)&as[r * 132 + c4 * 4] = v;
  }
  __syncthreads();

  const bool isQ = (v0 < LTOT);
  const float* W = isQ ? Wq : Wk;
  const float* bias = isQ ? bq : bk;

  for (int t = 0; t < 2; ++t) {
    const int n0 = (wave * 2 + t) * 16;
    const int n = n0 + m;          // this lane's output column
    v8f acc = {};
#pragma unroll
    for (int kk = 0; kk < 32; ++kk) {
      const int kofs = kk * 4 + hi * 2;
      v2f a = *(const v2f*)&as[m * 132 + kofs];
      v2f b;
      b.x = W[kofs * H + n];       // B[k][n] = W[k][n]
      b.y = W[(kofs + 1) * H + n];
      acc = wmma_f32(a, b, acc);
    }
    const float bv = bias[n];
#pragma unroll
    for (int r = 0; r < 8; ++r) {
      const int row = v0 + r + 8 * hi;
      const float val = acc[r] + bv;
      if (isQ) Qws[row * H + n] = val;
      else     Kws[(row - LTOT) * H + n] = val;
    }
  }
}

// ---------------------------------------------------------------------------
// Kernel 2: fused  P = Q@rel_emb^T  gather,  S = (Q@K^T + P[rel])/sqrt(H),
// split row-softmax -> m2c (384 cols), m2t (128 cols).
// One block = 16 query rows, 128 threads = 4 waves.
// ---------------------------------------------------------------------------
__global__ void __launch_bounds__(128)
attn_kernel(const float* __restrict__ Qws, const float* __restrict__ Kws,
            const int* __restrict__ relations,
            const float* __restrict__ rel_emb, float* __restrict__ out) {
  __shared__ __align__(16) float qs[16 * 132];   // Q tile
  __shared__ float ps[16 * 68];                  // P tile (cols padded)
  __shared__ float ss[16 * 516];                 // 16x512 scores, padded

  const int i0 = blockIdx.x * 16;
  const int tid = threadIdx.x;
  const int lane = tid & 31;
  const int wave = tid >> 5;
  const int m = lane & 15;
  const int hi = lane >> 4;

  for (int idx = tid; idx < 16 * 32; idx += 128) {
    const int r = idx >> 5, c4 = idx & 31;
    const float4 v = ((const float4*)(Qws + (size_t)(i0 + r) * H))[c4];
    *(float4*)&qs[r * 132 + c4 * 4] = v;
  }
  __syncthreads();

  // ---- P tile: wave w computes columns 16w .. 16w+15 (51 real rel ids) ----
  {
    const int n = wave * 16 + m;
    const float vf = (n < NREL) ? 1.0f : 0.0f;   // VALU mask, EXEC stays full
    const int nn = (n < NREL) ? n : 0;
    v8f acc = {};
#pragma unroll
    for (int kk = 0; kk < 32; ++kk) {
      const int kofs = kk * 4 + hi * 2;
      v2f a = *(const v2f*)&qs[m * 132 + kofs];
      v2f b = *(const v2f*)(rel_emb + nn * H + kofs);  // B[k][n]=rel_emb[n][k]
      b.x *= vf;
      b.y *= vf;
      acc = wmma_f32(a, b, acc);
    }
#pragma unroll
    for (int r = 0; r < 8; ++r) ps[(r + 8 * hi) * 68 + n] = acc[r];
  }
  __syncthreads();

  const float invs = 0.08838834764831845f;  // 1/sqrt(128)

  // ---- score tiles: wave w owns column tiles [w*128, w*128+128) ----
  for (int t = 0; t < 8; ++t) {
    const int n0 = (wave * 8 + t) * 16;
    const int n = n0 + m;          // combined kv column 0..511
    v8f acc = {};
#pragma unroll
    for (int kk = 0; kk < 32; ++kk) {
      const int kofs = kk * 4 + hi * 2;
      v2f a = *(const v2f*)&qs[m * 132 + kofs];
      v2f b = *(const v2f*)(Kws + (size_t)n * H + kofs);  // B[k][n]=K[n][k]
      acc = wmma_f32(a, b, acc);
    }
#pragma unroll
    for (int r = 0; r < 8; ++r) {
      const int mr = r + 8 * hi;
      const int irow = i0 + mr;
      const int ridx = relations[(size_t)irow * LTOT + LQ + n];
      ss[mr * 516 + n] = (acc[r] + ps[mr * 68 + ridx]) * invs;
    }
  }
  __syncthreads();

  // ---- split softmax, wave handles rows wave*4 .. wave*4+3 ----
  float* m2c = out;                                  // (1024, 384)
  float* m2t = out + (size_t)LTOT * LC;              // (1024, 128)
  for (int r = 0; r < 4; ++r) {
    const int mr = wave * 4 + r;
    const int irow = i0 + mr;
    const float* srow = &ss[mr * 516];

    // m2c over columns [0, 384)
    float mx = -INFINITY;
    for (int j = lane; j < LC; j += 32) mx = fmaxf(mx, srow[j]);
#pragma unroll
    for (int k = 16; k >= 1; k >>= 1) mx = fmaxf(mx, __shfl_xor(mx, k, 32));
    float sum = 0.0f;
    for (int j = lane; j < LC; j += 32) sum += __expf(srow[j] - mx);
#pragma unroll
    for (int k = 16; k >= 1; k >>= 1) sum += __shfl_xor(sum, k, 32);
    const float inv = 1.0f / sum;
    for (int j = lane; j < LC; j += 32)
      m2c[(size_t)irow * LC + j] = __expf(srow[j] - mx) * inv;

    // m2t over columns [384, 512)
    float mx2 = -INFINITY;
    for (int j = lane; j < LT; j += 32) mx2 = fmaxf(mx2, srow[LC + j]);
#pragma unroll
    for (int k = 16; k >= 1; k >>= 1) mx2 = fmaxf(mx2, __shfl_xor(mx2, k, 32));
    float sum2 = 0.0f;
    for (int j = lane; j < LT; j += 32) sum2 += __expf(srow[LC + j] - mx2);
#pragma unroll
    for (int k = 16; k >= 1; k >>= 1) sum2 += __shfl_xor(sum2, k, 32);
    const float inv2 = 1.0f / sum2;
    for (int j = lane; j < LT; j += 32)
      m2t[(size_t)irow * LT + j] = __expf(srow[LC + j] - mx2) * inv2;
  }
}

// ---------------------------------------------------------------------------
extern "C" void kernel_launch(void* const* d_in, const int* in_sizes, int n_in,
                              void* d_out, int out_size, void* d_ws,
                              size_t ws_size, hipStream_t stream) {
  (void)in_sizes; (void)n_in; (void)out_size; (void)ws_size;
  const float* q_enc = (const float*)d_in[1];
  const float* c_enc = (const float*)d_in[2];
  const float* t_enc = (const float*)d_in[3];
  const int* relations = (const int*)d_in[4];
  const float* Wq = (const float*)d_in[5];
  const float* bq = (const float*)d_in[6];
  const float* Wk = (const float*)d_in[7];
  const float* bk = (const float*)d_in[8];
  const float* rel_emb = (const float*)d_in[9];

  float* Qws = (float*)d_ws;                 // 1024 x 128
  float* Kws = Qws + (size_t)LTOT * H;       // 512  x 128

  proj_kernel<<<(LTOT + LKV) / 16, 128, 0, stream>>>(q_enc, c_enc, t_enc, Wq,
                                                     bq, Wk, bk, Qws, Kws);
  attn_kernel<<<LTOT / 16, 128, 0, stream>>>(Qws, Kws, relations, rel_emb,
                                             (float*)d_out);
}